// Dist_Conv2D_15238543966606
// MI455X (gfx1250) — compile-verified
//
#include <hip/hip_runtime.h>

// Dist-Conv2D (L-inf distance transform "convolution") for MI455X / gfx1250.
//
// out[b,o,h,w] = max_{c,i,j} |W[o, c*9+i*3+j] - x_edgepad[b,c,h+i-1,w+j-1]| + bias[o]
// B=4, C=16, H=W=64, OC=64, K=3, replicate pad 1.
//
// Roofline: ~5 MB HBM traffic (0.2 us @ 23.3 TB/s) vs ~302M sub+absmax VALU
// ops -> VALU bound. No bilinear structure => WMMA inapplicable; use the
// CDNA5 async global->LDS path (ASYNCcnt) for tile staging and keep the inner
// loop at exactly 2 VALU ops (v_sub_f32, v_max_f32 with |.| modifier,
// VOPD dual-issuable) per (k, pixel), with the x window register-cached
// across 16 output channels.

#define TH        8     // output rows per block tile
#define LR        10    // padded rows staged in LDS  (TH + 2)
#define LC        66    // padded cols staged in LDS  (64 + 2)
#define C_IN      16
#define OC_BLK    16    // output channels per block
#define NTHREADS  256

#define C_LDS_STRIDE_B   (LR * LC * 4)      // 2640 B between channel slices in LDS
#define C_GLB_STRIDE_B   (64 * 64 * 4)      // 16384 B between channel slices in x

__global__ __launch_bounds__(NTHREADS)
void dist_conv2d_kernel(const float* __restrict__ x,
                        const float* __restrict__ w,
                        const float* __restrict__ bias,
                        float* __restrict__ out) {
  __shared__ float tile[C_IN * LR * LC];   // 42,240 B of 320 KB LDS

  const int tid = threadIdx.x;
  const int h0  = blockIdx.x * TH;   // 0..56
  const int b   = blockIdx.y;        // 0..3
  const int oc0 = blockIdx.z * OC_BLK;

  // ---------------------------------------------------------------
  // Stage the replicate-padded x tile into LDS with CDNA5 async
  // loads (ASYNCcnt-tracked). Decompose (r,col) once per spatial
  // position (<=3 per thread), clamp for 'edge' padding, then walk
  // the 16 input channels with constant-stride address adds. The
  // instruction IOFFSET can't be used: it is added to BOTH the LDS
  // and global addresses, and the two channel strides differ.
  // ---------------------------------------------------------------
  const float* xb = x + (size_t)b * (C_IN * 64 * 64);
  for (int s = tid; s < LR * LC; s += NTHREADS) {   // 660 positions
    int r   = s / LC;
    int col = s - r * LC;
    int gh = h0 - 1 + r;  gh = gh < 0 ? 0 : (gh > 63 ? 63 : gh);
    int gw = col - 1;     gw = gw < 0 ? 0 : (gw > 63 ? 63 : gw);
    unsigned srcOff = (unsigned)((gh * 64 + gw) * 4);
    unsigned ldsOff = (unsigned)(size_t)(&tile[s]);  // low 32 bits = LDS byte addr
#pragma unroll
    for (int c = 0; c < C_IN; ++c) {
      unsigned lo = ldsOff + (unsigned)(c * C_LDS_STRIDE_B);
      unsigned go = srcOff + (unsigned)(c * C_GLB_STRIDE_B);
      asm volatile("global_load_async_to_lds_b32 %0, %1, %2"
                   :: "v"(lo), "v"(go), "s"(xb)
                   : "memory");
    }
  }
#if __has_builtin(__builtin_amdgcn_s_wait_asynccnt)
  __builtin_amdgcn_s_wait_asynccnt(0);
#else
  asm volatile("s_wait_asynccnt 0" ::: "memory");
#endif
  __syncthreads();

  // ---------------------------------------------------------------
  // Compute: each thread owns 2 adjacent pixels (hl, w0) (hl, w0+1)
  // and OC_BLK output channels. For each input channel, cache the
  // 3x4 padded x window in VGPRs and reuse it across all channels.
  // Weights are block-uniform -> scalar loads via constant cache.
  // ---------------------------------------------------------------
  const int hl = tid >> 5;           // 0..7
  const int w0 = (tid & 31) * 2;     // 0..62 (global col, even)

  float acc[OC_BLK][2];
#pragma unroll
  for (int oo = 0; oo < OC_BLK; ++oo) { acc[oo][0] = 0.0f; acc[oo][1] = 0.0f; }

  for (int c = 0; c < C_IN; ++c) {
    float xv[3][4];
#pragma unroll
    for (int dr = 0; dr < 3; ++dr) {
#pragma unroll
      for (int dc = 0; dc < 4; ++dc) {
        xv[dr][dc] = tile[(c * LR + hl + dr) * LC + (w0 + dc)];
      }
    }
#pragma unroll
    for (int oo = 0; oo < OC_BLK; ++oo) {
      const float* wp = w + (size_t)(oc0 + oo) * 144 + c * 9;  // uniform -> s_load
#pragma unroll
      for (int i = 0; i < 3; ++i) {
#pragma unroll
        for (int j = 0; j < 3; ++j) {
          float wv = wp[i * 3 + j];
          // v_sub_f32 + v_max_f32 with |src| modifier (2 VALU per k,pixel)
          acc[oo][0] = fmaxf(acc[oo][0], __builtin_fabsf(wv - xv[i][j]));
          acc[oo][1] = fmaxf(acc[oo][1], __builtin_fabsf(wv - xv[i][j + 1]));
        }
      }
    }
  }

  // ---------------------------------------------------------------
  // Epilogue: + bias, paired 8-byte stores.
  // ---------------------------------------------------------------
  const int gh = h0 + hl;
#pragma unroll
  for (int oo = 0; oo < OC_BLK; ++oo) {
    const int o = oc0 + oo;
    const float bv = bias[o];
    size_t oidx = ((((size_t)b * 64 + o) * 64 + gh) * 64) + w0;
    float2 v;
    v.x = acc[oo][0] + bv;
    v.y = acc[oo][1] + bv;
    *(float2*)(out + oidx) = v;
  }
}

extern "C" void kernel_launch(void* const* d_in, const int* in_sizes, int n_in,
                              void* d_out, int out_size, void* d_ws, size_t ws_size,
                              hipStream_t stream) {
  (void)in_sizes; (void)n_in; (void)out_size; (void)d_ws; (void)ws_size;
  const float* x    = (const float*)d_in[0];   // (4,16,64,64)
  const float* wgt  = (const float*)d_in[1];   // (64,144)
  const float* bias = (const float*)d_in[2];   // (64,1,1)
  float* out        = (float*)d_out;           // (4,64,64,64)

  dim3 grid(64 / TH, 4, 64 / OC_BLK);          // (8, 4, 4) = 128 blocks
  dist_conv2d_kernel<<<grid, NTHREADS, 0, stream>>>(x, wgt, bias, out);
}